// SupervisedContrastiveLoss_18562848653983
// MI455X (gfx1250) — compile-verified
//
#include <hip/hip_runtime.h>
#include <hip/hip_bf16.h>
#include <math.h>
#include <stdint.h>

typedef __attribute__((ext_vector_type(2))) float v2f;
typedef __attribute__((ext_vector_type(8))) float v8f;

#define B_ROWS     4096
#define DIMS       128
#define NROWS      8192                 // 2*B
#define NTILES     (NROWS / 16)         // 512 column tiles
#define TEMP_INV   (1.0f / 0.07f)
#define EPS_NORM   1e-12f
#define LDS_STRIDE 132                  // 128 + 4 dword pad -> conflict-free ds_load_b64

// ---------------------------------------------------------------------------
// Kernel 0: zero the (loss_sum, valid_count) accumulator in workspace.
// ---------------------------------------------------------------------------
__global__ void scl_init_accum(float* __restrict__ accum) {
    if (threadIdx.x < 2) accum[threadIdx.x] = 0.0f;
}

// ---------------------------------------------------------------------------
// Kernel 1: L2-normalize rows of concat(a, b) -> norm buffer [8192 x 128] f32.
// One wave32 per row; each lane handles a float4 (32 lanes * 4 = 128).
// ---------------------------------------------------------------------------
__global__ __launch_bounds__(256) void scl_normalize(const float* __restrict__ a,
                                                     const float* __restrict__ b,
                                                     float* __restrict__ out) {
    const int row  = (int)((blockIdx.x * blockDim.x + threadIdx.x) >> 5);
    const int lane = (int)(threadIdx.x & 31);
    if (row >= NROWS) return;
    const float* src = (row < B_ROWS) ? (a + (size_t)row * DIMS)
                                      : (b + (size_t)(row - B_ROWS) * DIMS);
    float4 v = ((const float4*)src)[lane];
    float ss = v.x * v.x + v.y * v.y + v.z * v.z + v.w * v.w;
    #pragma unroll
    for (int m = 16; m >= 1; m >>= 1) ss += __shfl_xor(ss, m, 32);
    const float inv = 1.0f / fmaxf(sqrtf(ss), EPS_NORM);
    v.x *= inv; v.y *= inv; v.z *= inv; v.w *= inv;
    ((float4*)(out + (size_t)row * DIMS))[lane] = v;
}

// ---------------------------------------------------------------------------
// Async global->LDS fill of one 16x128 f32 B-tile (padded stride in LDS).
// 512 x b128 transfers; every thread of the 256-thread block issues 2.
// ---------------------------------------------------------------------------
__device__ __forceinline__ void fill_tile_async(const float* __restrict__ emb,
                                                int jrow, uint32_t lds_base, int tid) {
    #pragma unroll
    for (int p = 0; p < 2; ++p) {
        const int i   = tid + p * 256;
        const int row = i >> 5;          // 0..15
        const int seg = i & 31;          // 0..31 float4 segments per row
        const uint64_t ga = (uint64_t)(uintptr_t)(emb + (size_t)(jrow + row) * DIMS + seg * 4);
        const uint32_t la = lds_base + (uint32_t)((row * LDS_STRIDE + seg * 4) * 4);
        asm volatile("global_load_async_to_lds_b128 %0, %1, off"
                     :: "v"(la), "v"(ga) : "memory");
    }
}

// ---------------------------------------------------------------------------
// Kernel 2: fused sim-GEMM (fp32 WMMA) + masked softmax-sum reductions.
// Block = 256 threads (8 waves) owns 128 rows (16 per wave). All waves sweep
// the full 8192 columns together, sharing a double-buffered LDS B tile that
// is streamed in with GLOBAL_LOAD_ASYNC_TO_LDS_B128 (ASYNCcnt-fenced).
// Numerics: rows are unit-norm, so the reference row-max is the diagonal
// sim_ii = 1/T; softmax shift-invariance lets us use the fixed shift 1/T,
// making all accumulations linear (no online rescale, no per-tile shuffles).
// ---------------------------------------------------------------------------
__global__ __launch_bounds__(256) void scl_main(const float* __restrict__ emb,
                                                const int*   __restrict__ labels,
                                                float* __restrict__ accum) {
    __shared__ float bsh[2][16 * LDS_STRIDE];   // double-buffered B tile (padded)
    __shared__ int   lab_sh[B_ROWS];

    const int tid  = (int)threadIdx.x;
    const int wave = tid >> 5;
    const int lane = tid & 31;
    const int half = lane >> 4;                 // K/M group select for WMMA layout
    const int hl   = lane & 15;
    const int wrow = (int)blockIdx.x * 128 + wave * 16;   // this wave's 16 rows

    // Stage labels once per block.
    for (int i = tid; i < B_ROWS; i += 256) lab_sh[i] = labels[i];

    // ---- A tile (16 rows x 128) in native WMMA A layout, kept in VGPRs ----
    v2f areg[32];
    {
        const float* arow = emb + (size_t)(wrow + hl) * DIMS;
        #pragma unroll
        for (int kk = 0; kk < 32; ++kk) {
            const int kb = kk * 4 + half * 2;
            areg[kk].x = arow[kb];
            areg[kk].y = arow[kb + 1];
        }
    }

    int rowg[8], rlab[8];
    #pragma unroll
    for (int r = 0; r < 8; ++r) {
        rowg[r] = wrow + r + 8 * half;                    // row held by C VGPR r
        rlab[r] = labels[rowg[r] & (B_ROWS - 1)];         // lab = concat(labels, labels)
    }

    float srow[8], prow[8];                               // per-lane linear partials
    #pragma unroll
    for (int r = 0; r < 8; ++r) { srow[r] = 0.f; prow[r] = 0.f; }

    const uint32_t lds0 = (uint32_t)(uintptr_t)&bsh[0][0];
    const uint32_t lds1 = (uint32_t)(uintptr_t)&bsh[1][0];

    // Prime the pipeline: tile 0 -> buffer 0.
    fill_tile_async(emb, 0, lds0, tid);
    asm volatile("s_wait_asynccnt 0" ::: "memory");
    __syncthreads();

    const int cbase = hl * LDS_STRIDE + half * 2;         // per-lane B base (dwords)

    for (int t = 0; t < NTILES; ++t) {
        const int j   = t * 16;
        const int buf = t & 1;

        // Kick off next tile's DMA into the other buffer (overlaps compute).
        if (t + 1 < NTILES)
            fill_tile_async(emb, j + 16, buf ? lds0 : lds1, tid);

        // ---- 16x16 score tile: 32 chained f32 WMMAs over K=128 ----
        v8f c = {};
        #pragma unroll
        for (int kk = 0; kk < 32; ++kk) {
            v2f b = *(const v2f*)&bsh[buf][cbase + kk * 4];   // ds_load_b64, bank-conflict-free
            c = __builtin_amdgcn_wmma_f32_16x16x4_f32(false, areg[kk],
                                                      false, b,
                                                      (short)0, c,
                                                      false, false);
        }

        // ---- masked exp accumulation (per-lane, no cross-lane work) ----
        const int col  = j + hl;
        const int lcol = lab_sh[col & (B_ROWS - 1)];
        #pragma unroll
        for (int r = 0; r < 8; ++r) {
            const float e    = __expf(c[r] * TEMP_INV - TEMP_INV);  // shift by analytic row max 1/T
            const bool  self = (col == rowg[r]);
            const bool  pos  = (!self) && (lcol == rlab[r]);
            srow[r] += self ? 0.f : e;     // denominator excludes self
            prow[r] += pos  ? e   : 0.f;   // positives exclude self
        }

        // Own DMA done before anyone touches the other buffer next iteration.
        if (t + 1 < NTILES)
            asm volatile("s_wait_asynccnt 0" ::: "memory");
        __syncthreads();
    }

    // ---- one-shot per-row reductions (16-lane halves), then wave atomics ----
    float lossAcc = 0.f, cntAcc = 0.f;
    #pragma unroll
    for (int r = 0; r < 8; ++r) {
        float S = srow[r], P = prow[r];
        #pragma unroll
        for (int m = 8; m >= 1; m >>= 1) {
            S += __shfl_xor(S, m, 16);
            P += __shfl_xor(P, m, 16);
        }
        const bool valid = (P > 0.0f);                    // exp > 0 => P>0 iff positives exist
        lossAcc += valid ? (__logf(S) - __logf(P)) : 0.f; // -log(P/S)
        cntAcc  += valid ? 1.0f : 0.0f;
    }
    lossAcc += __shfl_xor(lossAcc, 16, 32);               // fold the two halves (rows 0-7 + 8-15)
    cntAcc  += __shfl_xor(cntAcc, 16, 32);
    if (lane == 0) {
        atomicAdd(&accum[0], lossAcc);
        atomicAdd(&accum[1], cntAcc);
    }
}

// ---------------------------------------------------------------------------
// Kernel 3: final scalar.
// ---------------------------------------------------------------------------
__global__ void scl_finalize(const float* __restrict__ accum, float* __restrict__ out) {
    out[0] = accum[0] / fmaxf(accum[1], 1.0f);
}

extern "C" void kernel_launch(void* const* d_in, const int* in_sizes, int n_in,
                              void* d_out, int out_size, void* d_ws, size_t ws_size,
                              hipStream_t stream) {
    const float* ea     = (const float*)d_in[0];
    const float* eb     = (const float*)d_in[1];
    const int*   labels = (const int*)d_in[2];
    float*       out    = (float*)d_out;

    float* norm  = (float*)d_ws;                                    // 8192*128 f32 = 4 MB
    float* accum = (float*)((char*)d_ws + (size_t)NROWS * DIMS * sizeof(float));

    scl_init_accum<<<1, 32, 0, stream>>>(accum);
    scl_normalize<<<NROWS / 8, 256, 0, stream>>>(ea, eb, norm);
    scl_main<<<NROWS / 128, 256, 0, stream>>>(norm, labels, accum);
    scl_finalize<<<1, 1, 0, stream>>>(accum, out);
}